// ResBlock_77867757076595
// MI455X (gfx1250) — compile-verified
//
#include <hip/hip_runtime.h>
#include <hip/hip_bf16.h>
#include <math.h>

// ---------------------------------------------------------------------------
// Types for CDNA5 WMMA f32 16x16x4 fragments
// ---------------------------------------------------------------------------
typedef __attribute__((ext_vector_type(2))) float v2f;
typedef __attribute__((ext_vector_type(8))) float v8f;

#define C_MID 64
#define C_OUTC 128
#define HEADS 8
#define BN_EPS 1e-5f

// ---------------------------------------------------------------------------
// Generic f32 WMMA GEMM: Out[nrows, COUT] = X[nrows, CIN] * W[CIN, COUT] (+bias)
// Block = 128 threads = 4 wave32; each wave owns a 16-row stripe, block owns 64
// rows. Weight slice (CIN x 16) is staged in LDS per column tile.
// A fragment layout (ISA 7.12.2, 32-bit A 16x4): lanes 0-15 hold K=k0,k0+1 for
// M=lane; lanes 16-31 hold K=k0+2,k0+3. B 4x16 mirrors C/D row striping.
// ---------------------------------------------------------------------------
template <int CIN, int COUT>
__global__ __launch_bounds__(128) void wmma_gemm_f32(
    const float* __restrict__ X, const float* __restrict__ W,
    const float* __restrict__ bias, float* __restrict__ Out, int nrows) {
  __shared__ float wl[CIN * 16];

  const int lane = threadIdx.x & 31;
  const int wave = threadIdx.x >> 5;  // 0..3
  const int l16 = lane & 15;
  const int half = lane >> 4;  // 0 or 1
  const int row0 = blockIdx.x * 64 + wave * 16;
  int arow = row0 + l16;
  if (arow >= nrows) arow = nrows - 1;  // clamp: keep EXEC all-ones for WMMA
  const float* __restrict__ xrow = X + (size_t)arow * CIN;
  const bool full_tile = (row0 + 16 <= nrows);  // wave-uniform

  for (int nt = 0; nt < COUT / 16; ++nt) {
    // stage W[:, nt*16 .. nt*16+15] -> LDS, row-major CIN x 16
    for (int i = threadIdx.x; i < CIN * 16; i += 128) {
      const int k = i >> 4, c = i & 15;
      wl[i] = W[(size_t)k * COUT + nt * 16 + c];
    }
    __syncthreads();

    const float bcol = bias ? bias[nt * 16 + l16] : 0.0f;
    v8f acc;
#pragma unroll
    for (int r = 0; r < 8; ++r) acc[r] = bcol;  // bias folded into C

#pragma unroll 4
    for (int k0 = 0; k0 < CIN; k0 += 4) {
      const int ka = k0 + half * 2;
      v2f a, b;
      a.x = xrow[ka];
      a.y = xrow[ka + 1];
      b.x = wl[ka * 16 + l16];
      b.y = wl[(ka + 1) * 16 + l16];
      acc = __builtin_amdgcn_wmma_f32_16x16x4_f32(false, a, false, b, (short)0,
                                                  acc, false, false);
    }

    // store: VGPR r holds (row0 + r + half*8, nt*16 + l16).
    // Uniform fast path: only the single boundary block takes the guarded one.
    float* __restrict__ obase = Out + (size_t)row0 * COUT + nt * 16 + l16;
    if (full_tile) {
#pragma unroll
      for (int r = 0; r < 8; ++r)
        obase[(size_t)(r + half * 8) * COUT] = acc[r];
    } else {
#pragma unroll
      for (int r = 0; r < 8; ++r) {
        const int row = row0 + r + half * 8;
        if (row < nrows) Out[(size_t)row * COUT + nt * 16 + l16] = acc[r];
      }
    }
    __syncthreads();
  }
}

// ---------------------------------------------------------------------------
// Per-column batch statistics (biased var) -> mean[c], rstd[c]
// one 256-thread block per column
// ---------------------------------------------------------------------------
__global__ __launch_bounds__(256) void colstats_kernel(
    const float* __restrict__ X, int nrows, int ncols,
    float* __restrict__ mean, float* __restrict__ rstd) {
  __shared__ float s1[256];
  __shared__ float s2[256];
  const int c = blockIdx.x;
  const int tid = threadIdx.x;
  float sum = 0.f, sum2 = 0.f;
  for (int r = tid; r < nrows; r += 256) {
    const float v = X[(size_t)r * ncols + c];
    sum += v;
    sum2 += v * v;
  }
  s1[tid] = sum;
  s2[tid] = sum2;
  __syncthreads();
  for (int off = 128; off > 0; off >>= 1) {
    if (tid < off) {
      s1[tid] += s1[tid + off];
      s2[tid] += s2[tid + off];
    }
    __syncthreads();
  }
  if (tid == 0) {
    const float inv_n = 1.0f / (float)nrows;
    const float mu = s1[0] * inv_n;
    const float var = s2[0] * inv_n - mu * mu;
    mean[c] = mu;
    rstd[c] = rsqrtf(var + BN_EPS);
  }
}

// ---------------------------------------------------------------------------
// In-place BN (training stats) + ELU(alpha=1)
// ---------------------------------------------------------------------------
__global__ __launch_bounds__(256) void bn_elu_kernel(
    float* __restrict__ H, const float* __restrict__ gamma,
    const float* __restrict__ beta, const float* __restrict__ mean,
    const float* __restrict__ rstd, int total, int ncols) {
  const int idx = blockIdx.x * 256 + threadIdx.x;
  if (idx >= total) return;
  const int c = idx % ncols;
  const float v = gamma[c] * (H[idx] - mean[c]) * rstd[c] + beta[c];
  H[idx] = v > 0.0f ? v : expm1f(v);
}

// ---------------------------------------------------------------------------
// hU[n,h] = sum_c H[n,c] * U[c,h]   (64x8, tiny -> VALU)
// ---------------------------------------------------------------------------
__global__ __launch_bounds__(256) void hu_kernel(const float* __restrict__ H,
                                                 const float* __restrict__ U,
                                                 float* __restrict__ hU,
                                                 int n) {
  const int idx = blockIdx.x * 256 + threadIdx.x;
  if (idx >= n * HEADS) return;
  const int node = idx >> 3, h = idx & 7;
  const float* __restrict__ hr = H + (size_t)node * C_MID;
  float acc = 0.f;
#pragma unroll 8
  for (int c = 0; c < C_MID; ++c) acc = fmaf(hr[c], U[c * HEADS + h], acc);
  hU[idx] = acc;
}

// ---------------------------------------------------------------------------
// Self-loop contribution (diff == 0 -> q = softmax(conv_c), constant).
// Initializes num[n,c] and cnt[n] (workspace is poisoned by the harness).
// ---------------------------------------------------------------------------
__global__ __launch_bounds__(256) void selfloop_init_kernel(
    const float* __restrict__ Y, const float* __restrict__ cc,
    float* __restrict__ num, float* __restrict__ cnt, int n) {
  const int idx = blockIdx.x * 256 + threadIdx.x;
  if (idx >= n * C_MID) return;
  const int node = idx >> 6, c = idx & 63;
  float l[HEADS], m = -1e30f;
#pragma unroll
  for (int h = 0; h < HEADS; ++h) {
    l[h] = cc[h];
    m = fmaxf(m, l[h]);
  }
  float s = 0.f;
#pragma unroll
  for (int h = 0; h < HEADS; ++h) {
    l[h] = __expf(l[h] - m);
    s += l[h];
  }
  const float inv = 1.0f / s;
  const float* __restrict__ yrow = Y + (size_t)node * (HEADS * C_MID) + c;
  float acc = 0.f;
#pragma unroll
  for (int h = 0; h < HEADS; ++h) acc = fmaf(l[h] * inv, yrow[h * C_MID], acc);
  num[idx] = acc;
  if (c == 0) cnt[node] = 1.0f;
}

// ---------------------------------------------------------------------------
// Edge kernel: one wave32 per edge (grid-stride).
// Per edge: 8-head softmax of hU[dst]-hU[src]+c (hU rows fetched as float4
// pairs -> global_load_b128), then lane covers channels {2*lane, 2*lane+1} of
// the weighted combine of Y[src] (float2 gathers; Y is 123MB < 192MB L2, so
// gathers are L2-resident). Next edge's Y row is prefetched. Scatter via f32
// atomics into num[dst].
// ---------------------------------------------------------------------------
__global__ __launch_bounds__(256) void feast_edge_kernel(
    const int* __restrict__ src, const int* __restrict__ dst,
    const float* __restrict__ hU, const float* __restrict__ cc,
    const float* __restrict__ Y, float* __restrict__ num,
    float* __restrict__ cnt, int nedges) {
  const int lane = threadIdx.x & 31;
  const int wave = (blockIdx.x * 256 + threadIdx.x) >> 5;
  const int nwaves = (gridDim.x * 256) >> 5;

  float ccr[HEADS];
#pragma unroll
  for (int h = 0; h < HEADS; ++h) ccr[h] = cc[h];

  for (int e = wave; e < nedges; e += nwaves) {
    const int s = src[e];
    const int d = dst[e];

    // prefetch next iteration's Y gather row into cache
    const int en = e + nwaves;
    if (en < nedges) {
      const int sn = src[en];
      __builtin_prefetch(Y + (size_t)sn * (HEADS * C_MID) + 2 * lane, 0, 1);
    }

    const float4* __restrict__ hus4 = (const float4*)(hU + (size_t)s * HEADS);
    const float4* __restrict__ hud4 = (const float4*)(hU + (size_t)d * HEADS);
    const float4 s0 = hus4[0], s1 = hus4[1];
    const float4 d0 = hud4[0], d1 = hud4[1];
    float l[HEADS];
    l[0] = d0.x - s0.x + ccr[0];
    l[1] = d0.y - s0.y + ccr[1];
    l[2] = d0.z - s0.z + ccr[2];
    l[3] = d0.w - s0.w + ccr[3];
    l[4] = d1.x - s1.x + ccr[4];
    l[5] = d1.y - s1.y + ccr[5];
    l[6] = d1.z - s1.z + ccr[6];
    l[7] = d1.w - s1.w + ccr[7];
    float m = -1e30f;
#pragma unroll
    for (int h = 0; h < HEADS; ++h) m = fmaxf(m, l[h]);
    float ssum = 0.f;
#pragma unroll
    for (int h = 0; h < HEADS; ++h) {
      l[h] = __expf(l[h] - m);
      ssum += l[h];
    }
    const float inv = 1.0f / ssum;

    const float* __restrict__ yrow =
        Y + (size_t)s * (HEADS * C_MID) + 2 * lane;
    float ax = 0.f, ay = 0.f;
#pragma unroll
    for (int h = 0; h < HEADS; ++h) {
      const float2 y2 = *(const float2*)(yrow + h * C_MID);
      const float q = l[h] * inv;
      ax = fmaf(q, y2.x, ax);
      ay = fmaf(q, y2.y, ay);
    }
    float* np = num + (size_t)d * C_MID + 2 * lane;
    atomicAdd(np, ax);
    atomicAdd(np + 1, ay);
    if (lane == 0) atomicAdd(&cnt[d], 1.0f);
  }
}

// ---------------------------------------------------------------------------
// agg = num / cnt + conv_b   (in place)
// ---------------------------------------------------------------------------
__global__ __launch_bounds__(256) void finalize_kernel(
    float* __restrict__ num, const float* __restrict__ cnt,
    const float* __restrict__ bias, int n) {
  const int idx = blockIdx.x * 256 + threadIdx.x;
  if (idx >= n * C_MID) return;
  const int node = idx >> 6, c = idx & 63;
  num[idx] = num[idx] / cnt[node] + bias[c];
}

// ---------------------------------------------------------------------------
// launch
// ---------------------------------------------------------------------------
extern "C" void kernel_launch(void* const* d_in, const int* in_sizes, int n_in,
                              void* d_out, int out_size, void* d_ws,
                              size_t ws_size, hipStream_t stream) {
  const float* x = (const float*)d_in[0];         // [N,128]
  const int* edge_index = (const int*)d_in[1];    // [2,E]
  const float* fc1_w = (const float*)d_in[2];     // [128,64]
  const float* fc1_b = (const float*)d_in[3];     // [64]
  const float* bn1_g = (const float*)d_in[4];
  const float* bn1_b = (const float*)d_in[5];
  const float* conv_w = (const float*)d_in[6];    // [64,512]
  const float* conv_u = (const float*)d_in[7];    // [64,8]
  const float* conv_c = (const float*)d_in[8];    // [8]
  const float* conv_b = (const float*)d_in[9];    // [64]
  const float* bn2_g = (const float*)d_in[10];
  const float* bn2_b = (const float*)d_in[11];
  const float* fc2_w = (const float*)d_in[12];    // [64,128]
  const float* fc2_b = (const float*)d_in[13];    // [128]

  const int N = in_sizes[0] / C_OUTC;
  const int E = in_sizes[1] / 2;
  const int* e_src = edge_index;
  const int* e_dst = edge_index + E;

  // workspace carve-up (floats)
  float* f = (float*)d_ws;
  float* H = f;                      // [N,64]   fc1 out -> bn1+elu in place
  float* hU = H + (size_t)N * 64;    // [N,8]
  float* Y = hU + (size_t)N * 8;     // [N,512]  H @ conv_w
  float* num = Y + (size_t)N * 512;  // [N,64]   aggregate -> bn2+elu in place
  float* cnt = num + (size_t)N * 64; // [N]
  float* mean = cnt + N;             // [64]
  float* rstd = mean + 64;           // [64]

  const int gRows = (N + 63) / 64;
  const int gNC = (N * C_MID + 255) / 256;

  // 1) fc1 (WMMA) + bias
  wmma_gemm_f32<C_OUTC, C_MID>
      <<<gRows, 128, 0, stream>>>(x, fc1_w, fc1_b, H, N);
  // 2) bn1 stats + bn1/elu in place
  colstats_kernel<<<C_MID, 256, 0, stream>>>(H, N, C_MID, mean, rstd);
  bn_elu_kernel<<<gNC, 256, 0, stream>>>(H, bn1_g, bn1_b, mean, rstd,
                                         N * C_MID, C_MID);
  // 3) attention logits precompute hU = H @ conv_u
  hu_kernel<<<(N * HEADS + 255) / 256, 256, 0, stream>>>(H, conv_u, hU, N);
  // 4) Y = H @ conv_w (WMMA; big GEMM hoisted off the edges)
  wmma_gemm_f32<C_MID, HEADS * C_MID>
      <<<gRows, 128, 0, stream>>>(H, conv_w, nullptr, Y, N);
  // 5) self-loop init (also initializes num/cnt)
  selfloop_init_kernel<<<gNC, 256, 0, stream>>>(Y, conv_c, num, cnt, N);
  // 6) edge scatter (wave per edge)
  feast_edge_kernel<<<2048, 256, 0, stream>>>(e_src, e_dst, hU, conv_c, Y, num,
                                              cnt, E);
  // 7) mean aggregate + conv bias
  finalize_kernel<<<gNC, 256, 0, stream>>>(num, cnt, conv_b, N);
  // 8) bn2 stats + bn2/elu in place
  colstats_kernel<<<C_MID, 256, 0, stream>>>(num, N, C_MID, mean, rstd);
  bn_elu_kernel<<<gNC, 256, 0, stream>>>(num, bn2_g, bn2_b, mean, rstd,
                                         N * C_MID, C_MID);
  // 9) fc2 (WMMA) + bias -> d_out [N,128]
  wmma_gemm_f32<C_MID, C_OUTC>
      <<<gRows, 128, 0, stream>>>(num, fc2_w, fc2_b, (float*)d_out, N);
}